// LSTM_37357625540749
// MI455X (gfx1250) — compile-verified
//
#include <hip/hip_runtime.h>

#define SEQ 32768
#define HID 5
#define NG  20   // 4*HID gates
#define INP 8

typedef __attribute__((ext_vector_type(2))) float v2f;
typedef __attribute__((ext_vector_type(8))) float v8f;

// ---------------------------------------------------------------------------
// Kernel 1: xg[t][n] = sum_k embed[tokens[t]][k] * w_ih[n][k] + (b_ih[n]+b_hh[n])
// GEMM M=32768, K=8, N=20 via V_WMMA_F32_16X16X4_F32 (fp32-exact).
// One wave per 16-row M tile; two N tiles (N=0..15, 16..19); K=8 = 2 chunks of 4.
// ---------------------------------------------------------------------------
__global__ __launch_bounds__(256) void lstm_xg_wmma(
    const int*   __restrict__ tokens,
    const float* __restrict__ embed,
    const float* __restrict__ w_ih,
    const float* __restrict__ b_ih,
    const float* __restrict__ b_hh,
    float*       __restrict__ xg)
{
    const int lane  = threadIdx.x & 31;
    const int wave  = threadIdx.x >> 5;
    const int tile  = blockIdx.x * 8 + wave;   // 2048 tiles of 16 rows
    const int m0    = tile * 16;

    const int mr    = lane & 15;               // row (A) / col (B,C,D) within tile
    const int khalf = (lane >> 4) * 2;         // lanes 16-31 hold K+2,K+3

    // A operand (16x4 f32, 2 VGPRs): rows m0+mr, two K chunks {0..3} and {4..7}
    const int tok = tokens[m0 + mr];
    const float* er = embed + tok * INP;
    v2f a0, a1;
    a0.x = er[khalf + 0];     a0.y = er[khalf + 1];
    a1.x = er[4 + khalf + 0]; a1.y = er[4 + khalf + 1];

    #pragma unroll
    for (int nt = 0; nt < 2; ++nt) {
        const int  n  = nt * 16 + mr;
        const bool nv = (n < NG);
        v2f b0 = {0.f, 0.f}, b1 = {0.f, 0.f};
        if (nv) {
            const float* wr = w_ih + n * INP;   // w_ih is [20][8] row-major
            b0.x = wr[khalf + 0];     b0.y = wr[khalf + 1];
            b1.x = wr[4 + khalf + 0]; b1.y = wr[4 + khalf + 1];
        }
        v8f c = {};
        c = __builtin_amdgcn_wmma_f32_16x16x4_f32(false, a0, false, b0,
                                                  (short)0, c, false, false);
        c = __builtin_amdgcn_wmma_f32_16x16x4_f32(false, a1, false, b1,
                                                  (short)0, c, false, false);
        if (nv) {
            const float bias  = b_ih[n] + b_hh[n];
            const int   mbase = m0 + ((lane >> 4) * 8);  // M = r + 8*(lane>=16)
            #pragma unroll
            for (int r = 0; r < 8; ++r)
                xg[(mbase + r) * NG + n] = c[r] + bias;
        }
    }
}

// ---------------------------------------------------------------------------
// Kernel 2: the serial 32768-step LSTM scan. One wave; lanes 0..19 own one
// gate row each. Per step: 5-FMA dot vs broadcast h, one exp+rcp activation,
// one shuffle round to gather i/f/g/o per state row, c/h update, one shuffle
// round to broadcast new h. xg loads software-pipelined 2 deep (L2-resident).
// Epilogue: hT @ W_w.T + W_b, then @ W2_w.T + W2_b -> out[2].
// ---------------------------------------------------------------------------
__global__ __launch_bounds__(32) void lstm_scan(
    const float* __restrict__ xg,
    const float* __restrict__ w_hh,
    const float* __restrict__ W_w,
    const float* __restrict__ W_b,
    const float* __restrict__ W2_w,
    const float* __restrict__ W2_b,
    float*       __restrict__ out)
{
    const int  lane   = threadIdx.x;
    const bool active = lane < NG;
    const int  row    = lane % HID;                       // state row this lane tracks
    const bool isG    = (lane >= 2 * HID) && (lane < 3 * HID);  // g-gate -> tanh

    float w0 = 0.f, w1 = 0.f, w2 = 0.f, w3 = 0.f, w4 = 0.f;
    if (active) {
        const float* wr = w_hh + lane * HID;              // w_hh [20][5] row-major
        w0 = wr[0]; w1 = wr[1]; w2 = wr[2]; w3 = wr[3]; w4 = wr[4];
    }

    float h0 = 0.f, h1 = 0.f, h2 = 0.f, h3 = 0.f, h4 = 0.f;
    float cst = 0.f;                                      // c[row], replicated per row class

    // depth-2 prefetch of the xg stream
    float p0 = active ? xg[lane]      : 0.f;
    float p1 = active ? xg[NG + lane] : 0.f;

    for (int t = 0; t < SEQ; ++t) {
        const float cur = p0;
        p0 = p1;
        const int t2 = t + 2;
        p1 = (active && t2 < SEQ) ? xg[t2 * NG + lane] : 0.f;

        // gate pre-activation for this lane's row
        float g = cur + w0 * h0 + w1 * h1 + w2 * h2 + w3 * h3 + w4 * h4;

        // unified activation: sigmoid(x)=1/(1+e^-x), tanh(x)=1-2/(1+e^{2x})
        const float xin = isG ? (2.0f * g) : (-g);
        const float e   = __expf(xin);
        const float rc  = __builtin_amdgcn_rcpf(1.0f + e);
        const float act = isG ? (1.0f - 2.0f * rc) : rc;

        // gather i/f/g/o for this lane's state row (sources always in lanes 0..19)
        const float ai = __shfl(act, row,            32);
        const float af = __shfl(act, row + HID,      32);
        const float ag = __shfl(act, row + 2 * HID,  32);
        const float ao = __shfl(act, row + 3 * HID,  32);

        cst = af * cst + ai * ag;
        const float ec  = __expf(2.0f * cst);
        const float thc = 1.0f - 2.0f * __builtin_amdgcn_rcpf(1.0f + ec);
        const float hn  = ao * thc;

        // broadcast new h (valid copies live in lanes 0..4)
        h0 = __shfl(hn, 0, 32);
        h1 = __shfl(hn, 1, 32);
        h2 = __shfl(hn, 2, 32);
        h3 = __shfl(hn, 3, 32);
        h4 = __shfl(hn, 4, 32);
    }

    if (lane < 2) {
        const float h[HID] = {h0, h1, h2, h3, h4};
        float acc = W2_b[lane];
        #pragma unroll
        for (int j = 0; j < HID; ++j) {
            float f1 = W_b[j];
            #pragma unroll
            for (int k = 0; k < HID; ++k)
                f1 += W_w[j * HID + k] * h[k];
            acc += W2_w[lane * HID + j] * f1;
        }
        out[lane] = acc;
    }
}

// ---------------------------------------------------------------------------
extern "C" void kernel_launch(void* const* d_in, const int* in_sizes, int n_in,
                              void* d_out, int out_size, void* d_ws, size_t ws_size,
                              hipStream_t stream) {
    (void)in_sizes; (void)n_in; (void)out_size; (void)ws_size;
    const int*   tokens = (const int*)  d_in[0];
    const float* embed  = (const float*)d_in[1];
    const float* w_ih   = (const float*)d_in[2];
    const float* w_hh   = (const float*)d_in[3];
    const float* b_ih   = (const float*)d_in[4];
    const float* b_hh   = (const float*)d_in[5];
    const float* W_w    = (const float*)d_in[6];
    const float* W_b    = (const float*)d_in[7];
    const float* W2_w   = (const float*)d_in[8];
    const float* W2_b   = (const float*)d_in[9];
    float* outp = (float*)d_out;
    float* xg   = (float*)d_ws;   // SEQ * 20 floats = 2.62 MB scratch

    // 2048 M-tiles of 16 rows, 8 waves (one tile each) per 256-thread block
    lstm_xg_wmma<<<256, 256, 0, stream>>>(tokens, embed, w_ih, b_ih, b_hh, xg);
    // single-wave serial scan + tiny dense epilogue
    lstm_scan<<<1, 32, 0, stream>>>(xg, w_hh, W_w, W_b, W2_w, W2_b, outp);
}